// HGCL_34548716929761
// MI455X (gfx1250) — compile-verified
//
#include <hip/hip_runtime.h>
#include <hip/hip_bf16.h>
#include <math.h>

// ---------------------------------------------------------------------------
// HGCL forward for MI455X (gfx1250, wave32).
// fp32 end-to-end; dense GEMMs use V_WMMA_F32_16X16X4_F32 with the weight
// panel staged into LDS via gfx1250 async global->LDS copies.
// SpMM stages are HBM-bound scatter/gather with f32 global atomics.
// ---------------------------------------------------------------------------

typedef float v2f __attribute__((ext_vector_type(2)));
typedef float v8f __attribute__((ext_vector_type(8)));

#if defined(__gfx1250__)
#define GFX1250_DEV 1
#else
#define GFX1250_DEV 0
#endif

#if GFX1250_DEV && __has_builtin(__builtin_amdgcn_wmma_f32_16x16x4_f32)
#define USE_WMMA_F32 1
#else
#define USE_WMMA_F32 0
#endif

__device__ inline float wred_sum(float v) {
#pragma unroll
  for (int m = 16; m > 0; m >>= 1) v += __shfl_xor(v, m, 32);
  return v;
}
__device__ inline float wred_max(float v) {
#pragma unroll
  for (int m = 16; m > 0; m >>= 1) v = fmaxf(v, __shfl_xor(v, m, 32));
  return v;
}

// ---------------------------------------------------------------------------
// WMMA GEMM, templated on output width NCOL (64 or 256):
//   C[M x NCOL] = act(A[M x Kd] @ Wt[Kd x NCOL] + bias)
// Block = 128 threads (4 waves) = 4 consecutive 16-row tiles; the block
// cooperatively stages W in 64-row K-chunks into LDS (async DMA on gfx1250),
// each wave keeps all NCOL/16 accumulators so A is streamed exactly once.
// act: 0 = none, 1 = leaky relu (0.25), 2 = gate (C = gate * sigmoid(res))
// Requires: M % 16 == 0, Kd % 64 == 0 (true for all call sites).
// ---------------------------------------------------------------------------
template <int NCOL>
__global__ __launch_bounds__(128) void k_gemm_t(
    const float* __restrict__ A, const float* __restrict__ Wt,
    const float* __restrict__ bias, const float* gate, float* __restrict__ C,
    int M, int Kd, int act) {
  constexpr int NB = NCOL / 16;  // accumulators per wave
  constexpr int KC = 64;         // K-chunk staged in LDS
  __shared__ __align__(16) float sW[KC * NCOL];  // 16KB (NCOL=64) / 64KB (256)

  const int tid = threadIdx.x;
  const int wv = tid >> 5;
  const int lane = tid & 31;
  const int half = lane >> 4;  // 0: K pair {0,1}/rows +0..7 ; 1: {2,3}/rows +8..15
  const int l = lane & 15;
  const int m0 = (blockIdx.x * 4 + wv) * 16;
  const bool active = (m0 < M);  // wave-uniform

  v8f acc[NB] = {};
  for (int kc = 0; kc < Kd; kc += KC) {
    __syncthreads();  // previous chunk fully consumed
    // ---- stage W[kc : kc+KC][0 : NCOL] (contiguous) into LDS ----
#if GFX1250_DEV
    for (int idx = tid * 4; idx < KC * NCOL; idx += 512) {
      unsigned lds = (unsigned)(uintptr_t)(sW + idx);
      const float* g = Wt + (size_t)kc * NCOL + idx;
      asm volatile("global_load_async_to_lds_b128 %0, %1, off"
                   :: "v"(lds), "v"(g) : "memory");
    }
#if __has_builtin(__builtin_amdgcn_s_wait_asynccnt)
    __builtin_amdgcn_s_wait_asynccnt(0);
#else
    asm volatile("s_wait_asynccnt 0x0" ::: "memory");
#endif
#else
    for (int idx = tid * 4; idx < KC * NCOL; idx += 512) {
      float4 v = *reinterpret_cast<const float4*>(Wt + (size_t)kc * NCOL + idx);
      *reinterpret_cast<float4*>(sW + idx) = v;
    }
#endif
    __syncthreads();  // staged panel visible to all waves

    if (active) {
#if USE_WMMA_F32
      const float* Arow = A + (size_t)(m0 + l) * Kd + kc + 2 * half;
      for (int s = 0; s < KC / 4; ++s) {
        v2f a;  // A 16x4 frag: lanes0-15 K={k0,k0+1}, lanes16-31 K={k0+2,k0+3}
        a.x = Arow[4 * s];
        a.y = Arow[4 * s + 1];
        const float* srow = sW + (4 * s + 2 * half) * NCOL + l;
#pragma unroll
        for (int t = 0; t < NB; ++t) {
          v2f b;  // B 4x16 frag from LDS (mirrored layout)
          b.x = srow[t * 16];
          b.y = srow[t * 16 + NCOL];
          acc[t] = __builtin_amdgcn_wmma_f32_16x16x4_f32(
              false, a, false, b, (short)0, acc[t], false, false);
        }
      }
#else
#pragma unroll
      for (int t = 0; t < NB; ++t)
        for (int r = 0; r < 8; ++r) {
          const int row = m0 + r + 8 * half;
          const int col = t * 16 + l;
          float s = acc[t][r];
          for (int k = 0; k < KC; ++k)
            s += A[(size_t)row * Kd + kc + k] * sW[k * NCOL + col];
          acc[t][r] = s;
        }
#endif
    }
  }

  if (!active) return;
  // D layout: VGPR r -> row m0 + r + 8*half, col t*16 + (lane&15)
#pragma unroll
  for (int t = 0; t < NB; ++t) {
    const int col = t * 16 + l;
    const float bv = bias ? bias[col] : 0.f;
#pragma unroll
    for (int r = 0; r < 8; ++r) {
      const int row = m0 + r + 8 * half;
      float v = acc[t][r] + bv;
      if (act == 1) v = (v > 0.f) ? v : 0.25f * v;
      else if (act == 2) v = gate[(size_t)row * NCOL + col] * (1.f / (1.f + expf(-v)));
      C[(size_t)row * NCOL + col] = v;
    }
  }
}

// ---------------------------------------------------------------------------
// elementwise helpers
// ---------------------------------------------------------------------------
__global__ void k_fill(float* p, float v, long n) {
  long i = (long)blockIdx.x * blockDim.x + threadIdx.x;
  if (i < n) p[i] = v;
}
__global__ void k_scale(float* __restrict__ dst, const float* __restrict__ src,
                        float s, long n) {
  long i = (long)blockIdx.x * blockDim.x + threadIdx.x;
  if (i < n) dst[i] = src[i] * s;
}
__global__ void k_invsqrt(float* p, long n) {  // deg -> dis in place
  long i = (long)blockIdx.x * blockDim.x + threadIdx.x;
  if (i < n) {
    float v = p[i];
    p[i] = (v > 0.f) ? (1.f / sqrtf(v)) : 0.f;
  }
}
__global__ void k_avg_update(const float* __restrict__ x0,
                             const float* __restrict__ ui0,
                             float* __restrict__ a, float* __restrict__ b, long n) {
  long i = (long)blockIdx.x * blockDim.x + threadIdx.x;
  if (i < n) {
    float v = 0.5f * (x0[i] + ui0[i]);
    a[i] = v;
    b[i] = v;
  }
}
__global__ void k_concat3(float* __restrict__ dst, const float* __restrict__ a,
                          const float* __restrict__ b, const float* __restrict__ c,
                          int rows) {
  long i = (long)blockIdx.x * blockDim.x + threadIdx.x;
  long n = (long)rows * 192;
  if (i >= n) return;
  int r = (int)(i / 192), col = (int)(i % 192);
  float v;
  if (col < 64)       v = a[(size_t)r * 64 + col];
  else if (col < 128) v = b[(size_t)r * 64 + (col - 64)];
  else                v = c[(size_t)r * 64 + (col - 128)];
  dst[i] = v;
}

// ---------------------------------------------------------------------------
// degree / normalized scatter (SpMM) — the HBM-bound hot path
// ---------------------------------------------------------------------------
__global__ void k_deg(const int* __restrict__ row, const float* __restrict__ val,
                      float* deg, int E) {
  int e = blockIdx.x * blockDim.x + threadIdx.x;
  if (e < E) atomicAdd(&deg[row[e]], val[e]);
}
__global__ void k_deg_ui(const int* __restrict__ u, const int* __restrict__ i,
                         const float* __restrict__ val, float* deg, int E, int un) {
  int e = blockIdx.x * blockDim.x + threadIdx.x;
  if (e >= E) return;
  float v = val[e];
  atomicAdd(&deg[u[e]], v);
  atomicAdd(&deg[un + i[e]], v);
}
// out[r] += w * feat[c], w = val * (dis ? dis[r]*dis[c] : 1); 16 lanes/edge, float4
__global__ void k_scatter(const int* __restrict__ ra, const int* __restrict__ ca,
                          const float* __restrict__ val, const float* __restrict__ dis,
                          const float* __restrict__ feat, float* out,
                          int E, int roff, int coff) {
  long gid = (long)blockIdx.x * blockDim.x + threadIdx.x;
  int e = (int)(gid >> 4);
  if (e >= E) return;
  int j = ((int)gid & 15) * 4;
  int r = ra[e] + roff;
  int c = ca[e] + coff;
  float w = val[e];
  if (dis) w *= dis[r] * dis[c];
  const float4 f = *reinterpret_cast<const float4*>(feat + (size_t)c * 64 + j);
  float* o = out + (size_t)r * 64 + j;
  atomicAdd(o + 0, w * f.x);
  atomicAdd(o + 1, w * f.y);
  atomicAdd(o + 2, w * f.z);
  atomicAdd(o + 3, w * f.w);
}

// ---------------------------------------------------------------------------
// wave-per-row kernels
// ---------------------------------------------------------------------------
__global__ void k_add_l2n(float* __restrict__ acc, const float* __restrict__ x,
                          int rows) {  // 64-wide rows
  int wave = (int)((blockIdx.x * blockDim.x + threadIdx.x) >> 5);
  int lane = threadIdx.x & 31;
  if (wave >= rows) return;
  const float* xr = x + (size_t)wave * 64 + lane * 2;
  float a = xr[0], b = xr[1];
  float ss = wred_sum(a * a + b * b);
  float s = 1.f / fmaxf(sqrtf(ss), 1e-12f);
  float* ar = acc + (size_t)wave * 64 + lane * 2;
  ar[0] += a * s;
  ar[1] += b * s;
}
__global__ void k_l2n256(float* __restrict__ x, int rows) {  // 256-wide, in place
  int wave = (int)((blockIdx.x * blockDim.x + threadIdx.x) >> 5);
  int lane = threadIdx.x & 31;
  if (wave >= rows) return;
  float* xr = x + (size_t)wave * 256 + lane * 8;
  float v[8];
  float ss = 0.f;
#pragma unroll
  for (int j = 0; j < 8; ++j) { v[j] = xr[j]; ss += v[j] * v[j]; }
  ss = wred_sum(ss);
  float s = 1.f / fmaxf(sqrtf(ss), 1e-12f);
#pragma unroll
  for (int j = 0; j < 8; ++j) xr[j] = v[j] * s;
}
__global__ void k_colmean(const float* __restrict__ x, float* sums, int rows) {
  int j = threadIdx.x;  // 256 threads = 256 columns
  int r0 = blockIdx.x * 128;
  int r1 = min(r0 + 128, rows);
  float s = 0.f;
  for (int r = r0; r < r1; ++r) s += x[(size_t)r * 256 + j];
  atomicAdd(&sums[j], s);
}

// Fused tail: lw1 = softmax(m1 + mean1, over d), lw2 = softmax(m2 + mean2, over k)
// t[k] = sum_d emb[d]*lw1[d,k];  out[d] = emb[d] + sum_k t[k]*lw2[k,d]
__global__ void k_fuse(const float* __restrict__ emb, const float* __restrict__ m1,
                       const float* __restrict__ m2, const float* __restrict__ mean1,
                       const float* __restrict__ mean2, float invR,
                       float* __restrict__ out, int rows) {
  int wave = (int)((blockIdx.x * blockDim.x + threadIdx.x) >> 5);
  int lane = threadIdx.x & 31;
  if (wave >= rows) return;
  size_t r = (size_t)wave;
  int d0 = lane * 2, d1 = d0 + 1;
  const float* m1r = m1 + r * 256;  // (d,k): element d*4+k
  float a0[4], a1[4];
#pragma unroll
  for (int k = 0; k < 4; ++k) {
    a0[k] = m1r[d0 * 4 + k] + mean1[d0 * 4 + k] * invR;
    a1[k] = m1r[d1 * 4 + k] + mean1[d1 * 4 + k] * invR;
  }
  float e0 = emb[r * 64 + d0], e1 = emb[r * 64 + d1];
  float t[4];
#pragma unroll
  for (int k = 0; k < 4; ++k) {  // softmax over d (64, cross-lane)
    float mk = wred_max(fmaxf(a0[k], a1[k]));
    float x0 = expf(a0[k] - mk), x1 = expf(a1[k] - mk);
    float sk = wred_sum(x0 + x1);
    t[k] = wred_sum(e0 * x0 + e1 * x1) / sk;
  }
  const float* m2r = m2 + r * 256;  // (k,d): element k*64+d
  float b0[4], b1[4];
#pragma unroll
  for (int k = 0; k < 4; ++k) {
    b0[k] = m2r[k * 64 + d0] + mean2[k * 64 + d0] * invR;
    b1[k] = m2r[k * 64 + d1] + mean2[k * 64 + d1] * invR;
  }
  float mb0 = fmaxf(fmaxf(b0[0], b0[1]), fmaxf(b0[2], b0[3]));
  float mb1 = fmaxf(fmaxf(b1[0], b1[1]), fmaxf(b1[2], b1[3]));
  float s0 = 0.f, s1 = 0.f;
#pragma unroll
  for (int k = 0; k < 4; ++k) {
    b0[k] = expf(b0[k] - mb0); s0 += b0[k];
    b1[k] = expf(b1[k] - mb1); s1 += b1[k];
  }
  float o0 = e0, o1 = e1;
#pragma unroll
  for (int k = 0; k < 4; ++k) {
    o0 += t[k] * b0[k] / s0;
    o1 += t[k] * b1[k] / s1;
  }
  out[r * 64 + d0] = o0;
  out[r * 64 + d1] = o1;
}

// ---------------------------------------------------------------------------
// host driver
// ---------------------------------------------------------------------------
extern "C" void kernel_launch(void* const* d_in, const int* in_sizes, int n_in,
                              void* d_out, int out_size, void* d_ws, size_t ws_size,
                              hipStream_t stream) {
  (void)n_in; (void)out_size; (void)ws_size;
  const int* uu_row = (const int*)d_in[0];
  const int* uu_col = (const int*)d_in[1];
  const float* uu_val = (const float*)d_in[2];
  const int* ii_row = (const int*)d_in[3];
  const int* ii_col = (const int*)d_in[4];
  const float* ii_val = (const float*)d_in[5];
  const int* ui_u = (const int*)d_in[6];
  const int* ui_i = (const int*)d_in[7];
  const float* ui_val = (const float*)d_in[8];
  const float* user_emb = (const float*)d_in[9];
  const float* item_emb = (const float*)d_in[10];
  const float* gwu = (const float*)d_in[11];
  const float* gwub = (const float*)d_in[12];
  const float* gwi = (const float*)d_in[13];
  const float* gwib = (const float*)d_in[14];
  const float* meta_u_W = (const float*)d_in[15];
  const float* meta_u_b = (const float*)d_in[16];
  const float* meta_i_W = (const float*)d_in[17];
  const float* meta_i_b = (const float*)d_in[18];
  const float* mw[4][4];
  for (int m = 0; m < 4; ++m)
    for (int j = 0; j < 4; ++j) mw[m][j] = (const float*)d_in[19 + m * 4 + j];

  const int UN = 50000, ITN = 80000, NT = UN + ITN;  // 130000, all %16==0
  const int Euu = in_sizes[0], Eii = in_sizes[3], Eui = in_sizes[6];
  const long ND = (long)NT * 64;

  float* W = (float*)d_ws;
  size_t o = 0;
  float* uEb   = W + o; o += ND;                 // gated uE | iE   (layer input)
  float* uiEb  = W + o; o += ND;                 // concat uiE      (layer input)
  float* x0    = W + o; o += ND;                 // u0 | i0 scatter target
  float* ui0   = W + o; o += ND;                 // ui0 scatter target
  float* accx  = W + o; o += ND;                 // sum over layers (u|i)
  float* accui = W + o; o += ND;                 // sum over layers (ui)
  float* embb  = W + o; o += ND;                 // userEmb | itemEmb
  float* uiEmb = W + o; o += ND;                 // uiEmb mean
  float* neigh = W + o; o += ND;                 // uneigh | ineigh
  float* dis_uu = W + o; o += UN;
  float* dis_ii = W + o; o += ITN;
  float* dis_ui = W + o; o += NT;
  float* catb  = W + o; o += (size_t)NT * 192;   // concat3 input to meta GEMMs
  float* tb    = W + o; o += ND;                 // tu | ti
  float* hb    = W + o; o += (size_t)ITN * 256;  // MLP hidden (reused x4)
  float* mu1   = W + o; o += (size_t)UN * 256;
  float* mu2   = W + o; o += (size_t)UN * 256;
  float* mi1   = W + o; o += (size_t)ITN * 256;
  float* mi2   = W + o; o += (size_t)ITN * 256;
  float* meanb = W + o; o += 1024;               // 4 x 256 column sums

  auto cdiv = [](long a, long b) { return (int)((a + b - 1) / b); };
  auto fill = [&](float* p, float v, long n) {
    k_fill<<<cdiv(n, 256), 256, 0, stream>>>(p, v, n);
  };
  auto copy = [&](float* d, const float* s, long n) {
    k_scale<<<cdiv(n, 256), 256, 0, stream>>>(d, s, 1.f, n);
  };
  auto gemm = [&](const float* A, const float* Wt, const float* bias,
                  const float* gate, float* C, int M, int Kd, int Nc, int act) {
    int blocks = cdiv(M / 16, 4);  // 4 waves = 4 m-tiles per block
    if (Nc == 64)
      k_gemm_t<64><<<blocks, 128, 0, stream>>>(A, Wt, bias, gate, C, M, Kd, act);
    else
      k_gemm_t<256><<<blocks, 128, 0, stream>>>(A, Wt, bias, gate, C, M, Kd, act);
  };
  auto scatter = [&](const int* ra, const int* ca, const float* v, const float* dis,
                     const float* feat, float* out, int E, int roff, int coff) {
    k_scatter<<<cdiv((long)E * 16, 256), 256, 0, stream>>>(ra, ca, v, dis, feat, out,
                                                           E, roff, coff);
  };
  auto per_row = [&](int rows) { return cdiv((long)rows * 32, 256); };

  // ---- degrees (layer-invariant) -> inverse sqrt ----
  fill(dis_uu, 0.f, (long)(UN + ITN + NT));  // dis_uu|dis_ii|dis_ui contiguous
  k_deg<<<cdiv(Euu, 256), 256, 0, stream>>>(uu_row, uu_val, dis_uu, Euu);
  k_deg<<<cdiv(Eii, 256), 256, 0, stream>>>(ii_row, ii_val, dis_ii, Eii);
  k_deg_ui<<<cdiv(Eui, 256), 256, 0, stream>>>(ui_u, ui_i, ui_val, dis_ui, Eui, UN);
  k_invsqrt<<<cdiv((long)(UN + ITN + NT), 256), 256, 0, stream>>>(
      dis_uu, (long)(UN + ITN + NT));

  // ---- gates: uu0 = em * sigmoid(em@W + b) ----
  gemm(user_emb, gwu, gwub, user_emb, uEb, UN, 64, 64, 2);
  gemm(item_emb, gwi, gwib, item_emb, uEb + (size_t)UN * 64, ITN, 64, 64, 2);
  // uiE(layer0) = raw concat
  copy(uiEb, user_emb, (long)UN * 64);
  copy(uiEb + (size_t)UN * 64, item_emb, (long)ITN * 64);
  // layer-sum accumulators start at layer-0 entries
  copy(accx, uEb, ND);
  copy(accui, uiEb, ND);

  // ---- 2 GCN layers ----
  for (int layer = 0; layer < 2; ++layer) {
    fill(x0, 0.f, ND);
    fill(ui0, 0.f, ND);
    scatter(uu_row, uu_col, uu_val, dis_uu, uEb, x0, Euu, 0, 0);
    scatter(ii_row, ii_col, ii_val, dis_ii, uEb + (size_t)UN * 64,
            x0 + (size_t)UN * 64, Eii, 0, 0);
    scatter(ui_u, ui_i, ui_val, dis_ui, uiEb, ui0, Eui, 0, UN);  // u <- i
    scatter(ui_i, ui_u, ui_val, dis_ui, uiEb, ui0, Eui, UN, 0);  // i <- u
    k_add_l2n<<<per_row(NT), 256, 0, stream>>>(accx, x0, NT);
    k_add_l2n<<<per_row(NT), 256, 0, stream>>>(accui, ui0, NT);
    k_avg_update<<<cdiv(ND, 256), 256, 0, stream>>>(x0, ui0, uEb, uiEb, ND);
  }

  // ---- layer means ----
  k_scale<<<cdiv(ND, 256), 256, 0, stream>>>(embb, accx, 1.f / 3.f, ND);
  k_scale<<<cdiv(ND, 256), 256, 0, stream>>>(uiEmb, accui, 1.f / 3.f, ND);

  // ---- neighbor aggregation (plain spmm, no normalization) ----
  fill(neigh, 0.f, ND);
  scatter(ui_u, ui_i, ui_val, nullptr, uiEmb, neigh, Eui, 0, UN);  // uneigh
  scatter(ui_i, ui_u, ui_val, nullptr, uiEmb, neigh + (size_t)UN * 64, Eui, 0, 0);

  // ---- meta transforms: tu/ti = concat(emb, uiE, neigh) @ metaW + b ----
  k_concat3<<<cdiv((long)UN * 192, 256), 256, 0, stream>>>(catb, embb, uiEmb, neigh,
                                                           UN);
  k_concat3<<<cdiv((long)ITN * 192, 256), 256, 0, stream>>>(
      catb + (size_t)UN * 192, embb + (size_t)UN * 64, uiEmb + (size_t)UN * 64,
      neigh + (size_t)UN * 64, ITN);
  gemm(catb, meta_u_W, meta_u_b, nullptr, tb, UN, 192, 64, 0);
  gemm(catb + (size_t)UN * 192, meta_i_W, meta_i_b, nullptr,
       tb + (size_t)UN * 64, ITN, 192, 64, 0);

  // ---- 4 MLPs: leaky(x@Wp+bp) @ Wo + bo, then row l2-norm ----
  auto mlp = [&](const float* x, int rows, int mi, float* dst) {
    gemm(x, mw[mi][0], mw[mi][1], nullptr, hb, rows, 64, 256, 1);
    gemm(hb, mw[mi][2], mw[mi][3], nullptr, dst, rows, 256, 256, 0);
    k_l2n256<<<per_row(rows), 256, 0, stream>>>(dst, rows);
  };
  mlp(tb, UN, 0, mu1);
  mlp(tb, UN, 1, mu2);
  mlp(tb + (size_t)UN * 64, ITN, 2, mi1);
  mlp(tb + (size_t)UN * 64, ITN, 3, mi2);

  // ---- column sums for mean(0) ----
  fill(meanb, 0.f, 1024);
  k_colmean<<<cdiv(UN, 128), 256, 0, stream>>>(mu1, meanb, UN);
  k_colmean<<<cdiv(UN, 128), 256, 0, stream>>>(mu2, meanb + 256, UN);
  k_colmean<<<cdiv(ITN, 128), 256, 0, stream>>>(mi1, meanb + 512, ITN);
  k_colmean<<<cdiv(ITN, 128), 256, 0, stream>>>(mi2, meanb + 768, ITN);

  // ---- fused double-softmax + einsums + residual add ----
  float* outp = (float*)d_out;
  k_fuse<<<per_row(UN), 256, 0, stream>>>(embb, mu1, mu2, meanb, meanb + 256,
                                          1.f / (float)UN, outp, UN);
  k_fuse<<<per_row(ITN), 256, 0, stream>>>(embb + (size_t)UN * 64, mi1, mi2,
                                           meanb + 512, meanb + 768,
                                           1.f / (float)ITN,
                                           outp + (size_t)UN * 64, ITN);
}